// GIN_73830487818378
// MI455X (gfx1250) — compile-verified
//
#include <hip/hip_runtime.h>

#define NNODES 50000
#define NEDGES 800000
#define NGRAPH 256
#define DHID   64
#define BN_EPS 1e-5f
#define LDS_STR 68   // 16-row stage stride (pad: 68 % 64 = 4 banks) per wave

typedef __attribute__((ext_vector_type(2))) float v2f;
typedef __attribute__((ext_vector_type(8))) float v8f;

// ---------------------------------------------------------------- utilities
__global__ void zero_f32(float* __restrict__ p, int n) {
    int i = blockIdx.x * blockDim.x + threadIdx.x;
    if (i < n) p[i] = 0.0f;
}

__global__ void copy4(const float* __restrict__ src, float* __restrict__ dst, int n4) {
    int i = blockIdx.x * blockDim.x + threadIdx.x;
    if (i < n4) ((float4*)dst)[i] = ((const float4*)src)[i];
}

// --------------------------------------------- weighted scatter-add (GIN agg)
// agg[dst[e], :] += x[src[e], :] * w[e]; one thread = one edge x 4 features.
__global__ void scatter_edges(const float* __restrict__ x, float* __restrict__ agg,
                              const int* __restrict__ srcIdx, const int* __restrict__ dstIdx,
                              const float* __restrict__ ew, int nE, int logC) {
    int t = blockIdx.x * blockDim.x + threadIdx.x;
    int e = t >> logC;
    if (e >= nE) return;
    int c = (t & ((1 << logC) - 1)) << 2;      // feature base
    int d = 1 << (logC + 2);                   // row width (128 or 64)
    int s  = srcIdx[e];
    int dd = dstIdx[e];
    float w = ew[e];
    const float4 v = *(const float4*)(x + (size_t)s * d + c);
    float* out = agg + (size_t)dd * d + c;
    atomicAdd(out + 0, v.x * w);
    atomicAdd(out + 1, v.y * w);
    atomicAdd(out + 2, v.z * w);
    atomicAdd(out + 3, v.w * w);
}

// ------------------------------------------------------- fused MLP via WMMA
// Z = relu( relu(A @ W1 + b1) @ W2 + b2 ), A:[N,K], W1:[K,64], W2:[64,64]
// One wave per 16-row tile; fp32 WMMA 16x16x4 keeps exact reference numerics.
__global__ __launch_bounds__(256)
void mlp_wmma(const float* __restrict__ A, const float* __restrict__ W1,
              const float* __restrict__ b1, const float* __restrict__ W2,
              const float* __restrict__ b2, float* __restrict__ Z, int K) {
    __shared__ float lds[8][16 * LDS_STR];    // per-wave 16x64 staging (padded)
    const int wave = threadIdx.x >> 5;
    const int lane = threadIdx.x & 31;
    const int hi   = lane >> 4;               // half-wave select (K phase)
    const int ln   = lane & 15;
    const int tile = blockIdx.x * 8 + wave;
    const int m0   = tile * 16;
    if (m0 >= NNODES) return;                 // whole-wave exit: EXEC stays all-1s

    // --- GEMM1 accumulators seeded with b1 (bias is per-column == per-lane)
    v8f acc1[4];
    for (int nt = 0; nt < 4; ++nt) {
        float bv = b1[nt * 16 + ln];
        for (int j = 0; j < 8; ++j) acc1[nt][j] = bv;
    }
    const float* arow = A + (size_t)(m0 + ln) * K;
    for (int k = 0; k < K; k += 4) {
        int ka = k + hi * 2;
        v2f a = *(const v2f*)(arow + ka);      // A frag: lane row, K pair
        for (int nt = 0; nt < 4; ++nt) {
            const float* wp = W1 + (size_t)ka * DHID + nt * 16 + ln;
            v2f b; b.x = wp[0]; b.y = wp[DHID];
            acc1[nt] = __builtin_amdgcn_wmma_f32_16x16x4_f32(
                false, a, false, b, (short)0, acc1[nt], false, false);
        }
    }

    // --- ReLU + transpose D-layout -> A-layout through per-wave LDS region
    float* hl = lds[wave];
    for (int nt = 0; nt < 4; ++nt)
        for (int j = 0; j < 8; ++j) {
            float h = acc1[nt][j];
            h = h > 0.0f ? h : 0.0f;
            hl[(j + hi * 8) * LDS_STR + nt * 16 + ln] = h;
        }
    asm volatile("s_wait_dscnt 0" ::: "memory");  // per-wave DS in-order; drain

    // --- GEMM2 accumulators seeded with b2
    v8f acc2[4];
    for (int nt = 0; nt < 4; ++nt) {
        float bv = b2[nt * 16 + ln];
        for (int j = 0; j < 8; ++j) acc2[nt][j] = bv;
    }
    for (int k = 0; k < DHID; k += 4) {
        int ka = k + hi * 2;
        v2f a; a.x = hl[ln * LDS_STR + ka]; a.y = hl[ln * LDS_STR + ka + 1];
        for (int nt = 0; nt < 4; ++nt) {
            const float* wp = W2 + (size_t)ka * DHID + nt * 16 + ln;
            v2f b; b.x = wp[0]; b.y = wp[DHID];
            acc2[nt] = __builtin_amdgcn_wmma_f32_16x16x4_f32(
                false, a, false, b, (short)0, acc2[nt], false, false);
        }
    }

    // --- outer ReLU, store pre-BN activations
    for (int nt = 0; nt < 4; ++nt)
        for (int j = 0; j < 8; ++j) {
            float h = acc2[nt][j];
            h = h > 0.0f ? h : 0.0f;
            Z[(size_t)(m0 + j + hi * 8) * DHID + nt * 16 + ln] = h;
        }
}

// ------------------------------------------------------------ BN statistics
__global__ void bn_stats(const float* __restrict__ Z, float* __restrict__ ssum,
                         float* __restrict__ ssq) {
    int f = threadIdx.x & 63;
    int r = blockIdx.x * (blockDim.x >> 6) + (threadIdx.x >> 6);
    int stride = gridDim.x * (blockDim.x >> 6);
    float s = 0.0f, q = 0.0f;
    for (; r < NNODES; r += stride) {
        float v = Z[(size_t)r * DHID + f];
        s += v; q += v * v;
    }
    atomicAdd(&ssum[f], s);
    atomicAdd(&ssq[f], q);
}

__global__ void bn_finalize(const float* __restrict__ ssum, const float* __restrict__ ssq,
                            const float* __restrict__ gamma, const float* __restrict__ beta,
                            float* __restrict__ scale, float* __restrict__ shift) {
    int f = threadIdx.x;
    if (f >= DHID) return;
    float mean = ssum[f] * (1.0f / NNODES);
    float var  = ssq[f] * (1.0f / NNODES) - mean * mean;
    float inv  = rsqrtf(var + BN_EPS);
    float sc   = inv * gamma[f];
    scale[f] = sc;
    shift[f] = beta[f] - mean * sc;
}

// ---------------------------------------------- BN apply + global_add_pool
__global__ void bn_apply_pool(const float* __restrict__ Z, const float* __restrict__ scale,
                              const float* __restrict__ shift, const int* __restrict__ batch,
                              float* __restrict__ Hout, float* __restrict__ pool, int poolOff) {
    int t = blockIdx.x * blockDim.x + threadIdx.x;
    if (t >= NNODES * DHID) return;
    int n = t >> 6, f = t & 63;
    float v = Z[t] * scale[f] + shift[f];
    Hout[t] = v;
    atomicAdd(&pool[(size_t)batch[n] * 192 + poolOff + f], v);
}

// --------------------------------------------------------------------- host
extern "C" void kernel_launch(void* const* d_in, const int* in_sizes, int n_in,
                              void* d_out, int out_size, void* d_ws, size_t ws_size,
                              hipStream_t stream) {
    const float* x     = (const float*)d_in[0];
    const int*   ei    = (const int*)d_in[1];      // [2, E]
    const int*   batch = (const int*)d_in[2];
    const float* ew    = (const float*)d_in[3];
    const int* srcIdx = ei;
    const int* dstIdx = ei + NEDGES;

    float* out_h = (float*)d_out;                   // xs[-1]: N x 64
    float* pool  = (float*)d_out + NNODES * DHID;   // x_pool: G x 192

    // workspace layout
    float* agg  = (float*)d_ws;                     // N x 128
    float* z    = agg + (size_t)NNODES * 128;       // N x 64 (pre-BN)
    float* hb0  = z   + (size_t)NNODES * DHID;      // N x 64
    float* hb1  = hb0 + (size_t)NNODES * DHID;      // N x 64
    float* ssum = hb1 + (size_t)NNODES * DHID;      // 64
    float* ssq  = ssum + DHID;                      // 64
    float* scl  = ssq  + DHID;                      // 64
    float* shf  = scl  + DHID;                      // 64

    zero_f32<<<(NGRAPH * 192 + 255) / 256, 256, 0, stream>>>(pool, NGRAPH * 192);

    const float* hin = x;
    int K = 128;
    float* houts[3] = {hb0, hb1, out_h};

    for (int layer = 0; layer < 3; ++layer) {
        const float* W1 = (const float*)d_in[4 + 6 * layer];
        const float* B1 = (const float*)d_in[5 + 6 * layer];
        const float* W2 = (const float*)d_in[6 + 6 * layer];
        const float* B2 = (const float*)d_in[7 + 6 * layer];
        const float* GM = (const float*)d_in[8 + 6 * layer];
        const float* BT = (const float*)d_in[9 + 6 * layer];

        int n4 = NNODES * K / 4;
        copy4<<<(n4 + 255) / 256, 256, 0, stream>>>(hin, agg, n4);

        int logC = (K == 128) ? 5 : 4;             // feature chunks of 4
        long nt  = (long)NEDGES << logC;
        scatter_edges<<<(int)((nt + 255) / 256), 256, 0, stream>>>(
            hin, agg, srcIdx, dstIdx, ew, NEDGES, logC);

        int tiles  = NNODES / 16;                  // 3125 exactly
        int blocks = (tiles + 7) / 8;
        mlp_wmma<<<blocks, 256, 0, stream>>>(agg, W1, B1, W2, B2, z, K);

        zero_f32<<<1, 128, 0, stream>>>(ssum, 2 * DHID);   // ssum + ssq contiguous
        bn_stats<<<256, 256, 0, stream>>>(z, ssum, ssq);
        bn_finalize<<<1, 64, 0, stream>>>(ssum, ssq, GM, BT, scl, shf);

        bn_apply_pool<<<(NNODES * DHID + 255) / 256, 256, 0, stream>>>(
            z, scl, shf, batch, houts[layer], pool, layer * DHID);

        hin = houts[layer];
        K = DHID;
    }
    (void)in_sizes; (void)n_in; (void)out_size; (void)ws_size;
}